// HyperGraphNet_25718264168627
// MI455X (gfx1250) — compile-verified
//
#include <hip/hip_runtime.h>

// ---------------------------------------------------------------------------
// Types for WMMA
// ---------------------------------------------------------------------------
typedef float v2f __attribute__((ext_vector_type(2)));
typedef float v8f __attribute__((ext_vector_type(8)));

// ---------------------------------------------------------------------------
// Zero-fill (grid-stride)
// ---------------------------------------------------------------------------
__global__ void zero_f32(float* __restrict__ p, size_t n) {
  size_t i = (size_t)blockIdx.x * blockDim.x + threadIdx.x;
  size_t stride = (size_t)gridDim.x * blockDim.x;
  for (; i < n; i += stride) p[i] = 0.0f;
}

// ---------------------------------------------------------------------------
// Degree count:  dn[node_idx[i]] += 1 ; de[edge_idx[i]] += 1
// ---------------------------------------------------------------------------
__global__ void degree_count(const int* __restrict__ nidx,
                             const int* __restrict__ eidx,
                             float* __restrict__ dn, float* __restrict__ de,
                             int npairs) {
  int i = blockIdx.x * blockDim.x + threadIdx.x;
  if (i < npairs) {
    atomicAdd(&dn[nidx[i]], 1.0f);
    atomicAdd(&de[eidx[i]], 1.0f);
  }
}

__global__ void invert_deg(float* __restrict__ d, int n) {
  int i = blockIdx.x * blockDim.x + threadIdx.x;
  if (i < n) {
    float v = d[i];
    d[i] = (v > 0.0f) ? (1.0f / v) : 0.0f;
  }
}

// ---------------------------------------------------------------------------
// FP32 WMMA GEMM with LDS-staged B panel.
//   C[M,N] = A[M,K] @ B[K,N]   (row-major; M%16==0, N%64==0, K%4==0)
//
// Block = 128 threads (4 waves). Block tile = 64M x 64N; wave w owns rows
// [m0+16w, m0+16w+16) and all 64 columns via 4 v8f accumulators.
//
// B panel (K x 64) is staged once per block into LDS, pair-interleaved:
//   lds_b[kk*64 + n] = { B[2kk][n0+n], B[2kk+1][n0+n] }
// so a lane's B fragment (b.x=B[ka][n], b.y=B[ka+1][n], ka=k+2*half) is one
// contiguous ds_load_b64. A fragment is one contiguous global_load_b64.
//
// Frag layouts per ISA 7.12.2 (32-bit, wave32):
//   A 16x4 : lane l -> row M=(l&15); float2 at K = k + 2*(l>>4)
//   B 4x16 : lane l -> col N=(l&15); float2 at K = k + 2*(l>>4)
//   C 16x16: VGPR v -> row M = v + 8*(l>>4), col = l&15
// ---------------------------------------------------------------------------
__global__ __launch_bounds__(128) void gemm_wmma_f32_lds(
    const float* __restrict__ A, const float* __restrict__ B,
    float* __restrict__ C, int M, int N, int K, int mtiles) {
  extern __shared__ float2 lds_b[];           // [K/2][64] interleaved pairs
  const int tid = threadIdx.x;
  const int n0  = blockIdx.y * 64;
  const int KK  = K >> 1;                     // number of k-pairs

  // ---- stage B panel into LDS (pair-interleaved), float4 traffic ----
  for (int p = tid * 4; p < KK * 64; p += 128 * 4) {
    int kk = p >> 6;                          // pair row
    int n  = p & 63;                          // column within panel
    const float* brow0 = B + (size_t)(2 * kk) * N + n0 + n;
    const float* brow1 = brow0 + N;
    float4 r0 = *(const float4*)brow0;
    float4 r1 = *(const float4*)brow1;
    float2* dst = lds_b + p;
    *(float4*)(dst)     = make_float4(r0.x, r1.x, r0.y, r1.y);
    *(float4*)(dst + 2) = make_float4(r0.z, r1.z, r0.w, r1.w);
  }
  __syncthreads();

  const int wave = tid >> 5;
  const int lane = tid & 31;
  const int half = lane >> 4;                 // 0 or 1
  const int l15  = lane & 15;
  const int mt   = blockIdx.x * 4 + wave;     // this wave's M tile
  if (mt >= mtiles) return;                   // after the only barrier: safe
  const int m0 = mt * 16;

  const float* Arow = A + (size_t)(m0 + l15) * K;
  v8f acc0 = {}, acc1 = {}, acc2 = {}, acc3 = {};
  for (int k = 0; k < K; k += 4) {
    const int ka = k + half * 2;
    v2f a = *(const v2f*)(Arow + ka);         // one global b64
    const float2* bp = lds_b + (size_t)((k >> 1) + half) * 64 + l15;
    v2f b0 = *(const v2f*)(bp + 0);           // four ds b64
    v2f b1 = *(const v2f*)(bp + 16);
    v2f b2 = *(const v2f*)(bp + 32);
    v2f b3 = *(const v2f*)(bp + 48);
    acc0 = __builtin_amdgcn_wmma_f32_16x16x4_f32(false, a, false, b0, (short)0, acc0, false, false);
    acc1 = __builtin_amdgcn_wmma_f32_16x16x4_f32(false, a, false, b1, (short)0, acc1, false, false);
    acc2 = __builtin_amdgcn_wmma_f32_16x16x4_f32(false, a, false, b2, (short)0, acc2, false, false);
    acc3 = __builtin_amdgcn_wmma_f32_16x16x4_f32(false, a, false, b3, (short)0, acc3, false, false);
  }

  float* Crow = C + (size_t)(m0 + half * 8) * N + n0 + l15;
#pragma unroll
  for (int v = 0; v < 8; ++v) {
    Crow[(size_t)v * N +  0] = acc0[v];
    Crow[(size_t)v * N + 16] = acc1[v];
    Crow[(size_t)v * N + 32] = acc2[v];
    Crow[(size_t)v * N + 48] = acc3[v];
  }
}

// ---------------------------------------------------------------------------
// Scatter pass 1: for each incidence pair p: ef[e,:] += xl[n,:]
// One block per pair; indices are block-uniform (scalar loads); float4 reads.
// ---------------------------------------------------------------------------
__global__ __launch_bounds__(64) void scatter_node_to_edge(
    const float* __restrict__ xl, float* __restrict__ ef,
    const int* __restrict__ nidx, const int* __restrict__ eidx,
    int npairs, int N) {
  int pair = blockIdx.x;
  if (pair >= npairs) return;
  int n = nidx[pair];
  int e = eidx[pair];
  const float* src = xl + (size_t)n * N;
  float* dst = ef + (size_t)e * N;
  for (int c = threadIdx.x * 4; c < N; c += blockDim.x * 4) {
    float4 v = *(const float4*)(src + c);
    atomicAdd(dst + c + 0, v.x);
    atomicAdd(dst + c + 1, v.y);
    atomicAdd(dst + c + 2, v.z);
    atomicAdd(dst + c + 3, v.w);
  }
}

// ---------------------------------------------------------------------------
// Scatter pass 2: for each pair p: out[n,:] += Binv[e] * ef[e,:]
// ---------------------------------------------------------------------------
__global__ __launch_bounds__(64) void scatter_edge_to_node(
    const float* __restrict__ ef, const float* __restrict__ Binv,
    float* __restrict__ out,
    const int* __restrict__ nidx, const int* __restrict__ eidx,
    int npairs, int N) {
  int pair = blockIdx.x;
  if (pair >= npairs) return;
  int n = nidx[pair];
  int e = eidx[pair];
  float s = Binv[e];
  const float* src = ef + (size_t)e * N;
  float* dst = out + (size_t)n * N;
  for (int c = threadIdx.x * 4; c < N; c += blockDim.x * 4) {
    float4 v = *(const float4*)(src + c);
    atomicAdd(dst + c + 0, v.x * s);
    atomicAdd(dst + c + 1, v.y * s);
    atomicAdd(dst + c + 2, v.z * s);
    atomicAdd(dst + c + 3, v.w * s);
  }
}

// ---------------------------------------------------------------------------
// Epilogue: h[r,c] = relu(Dinv[r]*h[r,c] + bias[c])
// ---------------------------------------------------------------------------
__global__ void conv_epilogue(float* __restrict__ h,
                              const float* __restrict__ Dinv,
                              const float* __restrict__ bias,
                              int M, int N) {
  size_t i = (size_t)blockIdx.x * blockDim.x + threadIdx.x;
  size_t total = (size_t)M * N;
  if (i >= total) return;
  int r = (int)(i / N);
  int c = (int)(i % N);
  float v = h[i] * Dinv[r] + bias[c];
  h[i] = v > 0.0f ? v : 0.0f;
}

// ---------------------------------------------------------------------------
// Global mean pool, accumulation stage: gsum[b,:] += h[node,:]; cnt[b] += 1
// One block (256 threads) per node.
// ---------------------------------------------------------------------------
__global__ __launch_bounds__(256) void pool_accum(
    const float* __restrict__ h, const int* __restrict__ batch,
    float* __restrict__ gsum, float* __restrict__ cnt, int M, int N) {
  int node = blockIdx.x;
  if (node >= M) return;
  int b = batch[node];
  const float* src = h + (size_t)node * N;
  float* dst = gsum + (size_t)b * N;
  for (int c = threadIdx.x; c < N; c += blockDim.x) atomicAdd(dst + c, src[c]);
  if (threadIdx.x == 0) atomicAdd(&cnt[b], 1.0f);
}

// ---------------------------------------------------------------------------
// MLP head: per graph g: gm = gsum[g]/max(cnt,1); g2 = relu(gm@Wl1+bl1);
// out[g] = g2 @ Wl2 + bl2.  One block (128 threads) per graph. Nh=256, Nm=128.
// ---------------------------------------------------------------------------
__global__ __launch_bounds__(128) void mlp_head(
    const float* __restrict__ gsum, const float* __restrict__ cnt,
    const float* __restrict__ Wl1, const float* __restrict__ bl1,
    const float* __restrict__ Wl2, const float* __restrict__ bl2,
    float* __restrict__ out, int G, int Nh, int Nm) {
  __shared__ float gm[256];
  __shared__ float red[128];
  int g = blockIdx.x;
  if (g >= G) return;
  int t = threadIdx.x;

  float invc = 1.0f / fmaxf(cnt[g], 1.0f);
  for (int c = t; c < Nh; c += blockDim.x) gm[c] = gsum[(size_t)g * Nh + c] * invc;
  __syncthreads();

  float acc = bl1[t];
  for (int k = 0; k < Nh; ++k) acc = fmaf(gm[k], Wl1[(size_t)k * Nm + t], acc);
  float h = fmaxf(acc, 0.0f);

  red[t] = h * Wl2[t];
  __syncthreads();
  for (int s = 64; s > 0; s >>= 1) {
    if (t < s) red[t] += red[t + s];
    __syncthreads();
  }
  if (t == 0) out[g] = red[0] + bl2[0];
}

// ---------------------------------------------------------------------------
// Host-side launcher
// ---------------------------------------------------------------------------
static inline size_t align256(size_t x) { return (x + 255) & ~(size_t)255; }

extern "C" void kernel_launch(void* const* d_in, const int* in_sizes, int n_in,
                              void* d_out, int out_size, void* d_ws, size_t ws_size,
                              hipStream_t stream) {
  (void)n_in; (void)ws_size;

  // ---- inputs (setup_inputs dict order) ----
  const float* x     = (const float*)d_in[0];   // [M, F]
  const int*   hidx  = (const int*)  d_in[1];   // [2, P]
  const int*   batch = (const int*)  d_in[2];   // [M]
  const float* W1    = (const float*)d_in[3];   // [F, H1]
  const float* b1    = (const float*)d_in[4];   // [H1]
  const float* W2    = (const float*)d_in[5];   // [H1, H2]
  const float* b2    = (const float*)d_in[6];   // [H2]
  const float* W3    = (const float*)d_in[7];   // [H2, H3]
  const float* b3    = (const float*)d_in[8];   // [H3]
  const float* Wl1   = (const float*)d_in[9];   // [H3, H3/2]
  const float* bl1   = (const float*)d_in[10];  // [H3/2]
  const float* Wl2   = (const float*)d_in[11];  // [H3/2, 1]
  const float* bl2   = (const float*)d_in[12];  // [1]
  float* out = (float*)d_out;

  const int M  = in_sizes[2];               // 50000 nodes
  const int P  = in_sizes[1] / 2;           // 400000 incidence pairs
  const int F  = in_sizes[0] / M;           // 128
  const int H1 = in_sizes[4];               // 256
  const int H2 = in_sizes[6];               // 512
  const int H3 = in_sizes[8];               // 256
  const int Hm = in_sizes[10];              // 128
  const int G  = out_size;                  // 1024 graphs
  const int E  = M;                         // N_EDGES == N_NODES == 50000

  const int* nidx = hidx;                   // row 0: node indices
  const int* eidx = hidx + P;               // row 1: edge indices

  // ---- workspace carve-out ----
  char* ws = (char*)d_ws;
  size_t off = 0;
  float* Dinv = (float*)(ws + off); off = align256(off + (size_t)M * 4);
  float* Binv = (float*)(ws + off); off = align256(off + (size_t)E * 4);
  float* hA   = (float*)(ws + off); off = align256(off + (size_t)M * H1 * 4); // width<=256
  float* hB   = (float*)(ws + off); off = align256(off + (size_t)M * H2 * 4); // width<=512
  float* xl   = (float*)(ws + off); off = align256(off + (size_t)M * H2 * 4);
  float* ef   = (float*)(ws + off); off = align256(off + (size_t)M * H2 * 4);
  float* gsum = (float*)(ws + off); off = align256(off + (size_t)G * H3 * 4);
  float* cnt  = (float*)(ws + off); off = align256(off + (size_t)G * 4);

  const int ZB = 256;
  const int zgrid = 4096;   // grid-stride zero fill

  // ---- inverse degrees ----
  zero_f32<<<zgrid, ZB, 0, stream>>>(Dinv, (size_t)M);
  zero_f32<<<zgrid, ZB, 0, stream>>>(Binv, (size_t)E);
  degree_count<<<(P + ZB - 1) / ZB, ZB, 0, stream>>>(nidx, eidx, Dinv, Binv, P);
  invert_deg<<<(M + ZB - 1) / ZB, ZB, 0, stream>>>(Dinv, M);
  invert_deg<<<(E + ZB - 1) / ZB, ZB, 0, stream>>>(Binv, E);

  // ---- one hypergraph-conv layer ----
  auto layer = [&](const float* in, int K, const float* W, const float* bias,
                   int Nout, float* outbuf) {
    // xl = in @ W     (WMMA fp32, LDS-staged B panel)
    const int mtiles = M / 16;
    dim3 ggrid((mtiles + 3) / 4, Nout / 64);
    size_t shmem = (size_t)(K / 2) * 64 * sizeof(float2);   // K*256 bytes
    gemm_wmma_f32_lds<<<ggrid, 128, shmem, stream>>>(in, W, xl, M, Nout, K, mtiles);
    // clear edge features and output accumulator
    zero_f32<<<zgrid, ZB, 0, stream>>>(ef, (size_t)E * Nout);
    zero_f32<<<zgrid, ZB, 0, stream>>>(outbuf, (size_t)M * Nout);
    // node -> edge aggregation
    scatter_node_to_edge<<<P, 64, 0, stream>>>(xl, ef, nidx, eidx, P, Nout);
    // edge -> node aggregation (B^-1 folded in)
    scatter_edge_to_node<<<P, 64, 0, stream>>>(ef, Binv, outbuf, nidx, eidx, P, Nout);
    // D^-1 scale + bias + relu
    size_t total = (size_t)M * Nout;
    conv_epilogue<<<(unsigned)((total + ZB - 1) / ZB), ZB, 0, stream>>>(
        outbuf, Dinv, bias, M, Nout);
  };

  layer(x,  F,  W1, b1, H1, hA);   // 128 -> 256
  layer(hA, H1, W2, b2, H2, hB);   // 256 -> 512
  layer(hB, H2, W3, b3, H3, hA);   // 512 -> 256

  // ---- global mean pool + MLP head ----
  zero_f32<<<zgrid, ZB, 0, stream>>>(gsum, (size_t)G * H3);
  zero_f32<<<zgrid, ZB, 0, stream>>>(cnt, (size_t)G);
  pool_accum<<<M, 256, 0, stream>>>(hA, batch, gsum, cnt, M, H3);
  mlp_head<<<G, 128, 0, stream>>>(gsum, cnt, Wl1, bl1, Wl2, bl2, out, G, H3, Hm);
}